// SingleGraphWrapper_71305047048704
// MI455X (gfx1250) — compile-verified
//
#include <hip/hip_runtime.h>

typedef __attribute__((ext_vector_type(2))) float v2f;
typedef __attribute__((ext_vector_type(8))) float v8f;

#define FDIM 64
#define HDIM 64

// ---------------- degree / normalization ----------------

__global__ void deg_init_kernel(float* deg, int N) {
  int i = blockIdx.x * blockDim.x + threadIdx.x;
  if (i < N) deg[i] = 1.0f;  // self-loop weight
}

__global__ void deg_acc_kernel(const int* __restrict__ row,
                               const float* __restrict__ attr,
                               float* deg, int E) {
  int e = blockIdx.x * blockDim.x + threadIdx.x;
  if (e < E) atomicAdd(&deg[row[e]], attr[e]);
}

__global__ void dinv_kernel(float* deg, int N) {
  int i = blockIdx.x * blockDim.x + threadIdx.x;
  if (i < N) {
    float d = deg[i];
    deg[i] = (d > 0.0f) ? rsqrtf(d) : 0.0f;
  }
}

__global__ void norm_kernel(const int* __restrict__ row,
                            const int* __restrict__ col,
                            const float* __restrict__ attr,
                            const float* __restrict__ dinv,
                            float* __restrict__ nrm, int E) {
  int e = blockIdx.x * blockDim.x + threadIdx.x;
  if (e < E) nrm[e] = dinv[row[e]] * attr[e] * dinv[col[e]];
}

// ---------------- dense GEMM: Out[N,64] = A[N,64] @ W[64,64] (fp32 WMMA) ----
// W staged in LDS K-pair interleaved: sW2[k2*HDIM + n] = {W[2k2][n], W[2k2+1][n]}
// so each B fragment is one aligned ds_load_b64 straight into an even VGPR pair.

__global__ __launch_bounds__(128) void gemm_wmma_kernel(
    const float* __restrict__ A, const float* __restrict__ W,
    float* __restrict__ Out, int N) {
  __shared__ v2f sW2[(FDIM / 2) * HDIM];  // 16 KB
  int tid = threadIdx.x;
  for (int i = tid; i < (FDIM / 2) * HDIM; i += 128) {
    int k2 = i >> 6;          // K pair index
    int n  = i & (HDIM - 1);  // output column
    v2f p;
    p.x = W[(2 * k2)     * HDIM + n];
    p.y = W[(2 * k2 + 1) * HDIM + n];
    sW2[i] = p;
  }
  __syncthreads();

  int wave = tid >> 5;
  int lane = tid & 31;
  int m    = lane & 15;   // row within 16-row tile (A) / col within 16-col tile (B,C)
  int kh   = lane >> 4;   // K half-select per ISA layout

  long tile = (long)blockIdx.x * 4 + wave;
  long row0 = tile * 16;
  if (row0 >= N) return;  // uniform per-wave exit (no further barriers)

  v8f acc0 = {}, acc1 = {}, acc2 = {}, acc3 = {};

  for (int k4 = 0; k4 < FDIM; k4 += 4) {
    long r = row0 + m;
    if (r >= N) r = N - 1;  // clamp tail reads; stores are guarded below
    // A frag 16x4: v0 = K(k4+2*kh), v1 = K(k4+2*kh+1)  (lanes 0-15: K0/K1, 16-31: K2/K3)
    v2f a = *(const v2f*)&A[r * FDIM + k4 + 2 * kh];
    // B frag: {W[k][n], W[k+1][n]} with k = k4 + 2*kh, one b64 LDS load per tile
    const v2f* wrow = &sW2[(k4 / 2 + kh) * HDIM];
    acc0 = __builtin_amdgcn_wmma_f32_16x16x4_f32(false, a, false, wrow[m],      (short)0, acc0, false, false);
    acc1 = __builtin_amdgcn_wmma_f32_16x16x4_f32(false, a, false, wrow[16 + m], (short)0, acc1, false, false);
    acc2 = __builtin_amdgcn_wmma_f32_16x16x4_f32(false, a, false, wrow[32 + m], (short)0, acc2, false, false);
    acc3 = __builtin_amdgcn_wmma_f32_16x16x4_f32(false, a, false, wrow[48 + m], (short)0, acc3, false, false);
  }

  // C/D layout: VGPR j, lanes 0-15 -> M=j, lanes 16-31 -> M=j+8; col = lane&15
#pragma unroll
  for (int j = 0; j < 8; ++j) {
    long rr = row0 + j + 8 * kh;
    if (rr < N) {
      float* o = &Out[rr * HDIM];
      o[m]      = acc0[j];
      o[16 + m] = acc1[j];
      o[32 + m] = acc2[j];
      o[48 + m] = acc3[j];
    }
  }
}

// ---------------- aggregation ----------------

// agg[n,:] = hW[n,:] * dinv[n]^2   (self-loop contribution; also zero-initializes)
__global__ void agg_init_kernel(const float* __restrict__ hW,
                                const float* __restrict__ dinv,
                                float* __restrict__ agg, int N) {
  long i = (long)blockIdx.x * blockDim.x + threadIdx.x;  // over N*16 float4's
  if (i >= (long)N * 16) return;
  long n = i >> 4;
  float s = dinv[n]; s = s * s;
  float4 v = ((const float4*)hW)[i];
  v.x *= s; v.y *= s; v.z *= s; v.w *= s;
  ((float4*)agg)[i] = v;
}

// agg[row[e],:] += hW[col[e],:] * norm[e]
__global__ void agg_edge_kernel(const int* __restrict__ row,
                                const int* __restrict__ col,
                                const float* __restrict__ nrm,
                                const float* __restrict__ hW,
                                float* __restrict__ agg, int E) {
  long i = (long)blockIdx.x * blockDim.x + threadIdx.x;  // over E*16
  if (i >= (long)E * 16) return;
  int e  = (int)(i >> 4);
  int f4 = (int)(i & 15);
  float w = nrm[e];
  float4 v = ((const float4*)hW)[(long)col[e] * 16 + f4];
  float* dst = &agg[(long)row[e] * HDIM + f4 * 4];
  atomicAdd(dst + 0, v.x * w);
  atomicAdd(dst + 1, v.y * w);
  atomicAdd(dst + 2, v.z * w);
  atomicAdd(dst + 3, v.w * w);
}

// agg = relu(agg + bias)
__global__ void bias_relu_kernel(float* agg, const float* __restrict__ bias, int N) {
  long i = (long)blockIdx.x * blockDim.x + threadIdx.x;
  if (i >= (long)N * HDIM) return;
  float v = agg[i] + bias[i & (HDIM - 1)];
  agg[i] = v > 0.0f ? v : 0.0f;
}

// ---------------- pooling + classifier head ----------------

__global__ void pool_init_kernel(float* pooled) {
  pooled[threadIdx.x] = 0.0f;
}

__global__ __launch_bounds__(256) void pool_kernel(const float* __restrict__ h,
                                                   float* pooled, int N, int rowsPerBlock) {
  __shared__ float red[256];
  int f = threadIdx.x & 63;
  int g = threadIdx.x >> 6;  // 0..3
  long r0 = (long)blockIdx.x * rowsPerBlock;
  long r1 = r0 + rowsPerBlock; if (r1 > N) r1 = N;
  float s = 0.0f;
  for (long r = r0 + g; r < r1; r += 4) s += h[r * HDIM + f];
  red[threadIdx.x] = s;
  __syncthreads();
  if (g == 0) {
    float t = red[f] + red[64 + f] + red[128 + f] + red[192 + f];
    atomicAdd(&pooled[f], t);
  }
}

__global__ __launch_bounds__(64) void head_kernel(
    const float* __restrict__ pooled, const float* __restrict__ h_other,
    const float* __restrict__ Wc1, const float* __restrict__ bc1,
    const float* __restrict__ Wc2, const float* __restrict__ bc2,
    float* __restrict__ out, int N) {
  __shared__ float z[2 * HDIM];
  __shared__ float hid[HDIM];
  int t = threadIdx.x;
  z[t]        = pooled[t] * (1.0f / (float)N);  // global mean pool
  z[HDIM + t] = h_other[t];
  __syncthreads();
  float acc = bc1[t];
  for (int i = 0; i < 2 * HDIM; ++i) acc += z[i] * Wc1[i * HDIM + t];
  hid[t] = acc > 0.0f ? acc : 0.0f;
  __syncthreads();
  if (t < 3) {
    float o = bc2[t];
    for (int i = 0; i < HDIM; ++i) o += hid[i] * Wc2[i * 3 + t];
    out[t] = o;
  }
}

// ---------------- launch ----------------

extern "C" void kernel_launch(void* const* d_in, const int* in_sizes, int n_in,
                              void* d_out, int out_size, void* d_ws, size_t ws_size,
                              hipStream_t stream) {
  (void)n_in; (void)out_size; (void)ws_size;

  const float* x         = (const float*)d_in[0];
  const int*   edge_idx  = (const int*)  d_in[1];
  const float* edge_attr = (const float*)d_in[2];
  /* d_in[3] = batch: all zeros, single graph -> pool over all N */
  const float* W1  = (const float*)d_in[4];
  const float* b1  = (const float*)d_in[5];
  const float* W2  = (const float*)d_in[6];
  const float* b2  = (const float*)d_in[7];
  const float* Wc1 = (const float*)d_in[8];
  const float* bc1 = (const float*)d_in[9];
  const float* Wc2 = (const float*)d_in[10];
  const float* bc2 = (const float*)d_in[11];
  const float* h_other = (const float*)d_in[12];

  int N = in_sizes[0] / FDIM;
  int E = in_sizes[2];
  const int* row = edge_idx;       // edge_index[0,:]
  const int* col = edge_idx + E;   // edge_index[1,:]

  // workspace carve-up (all buffers fully written before read each call)
  char* ws = (char*)d_ws;
  size_t off = 0;
  auto carve = [&](size_t bytes) {
    char* p = ws + off;
    off += (bytes + 255) & ~(size_t)255;
    return p;
  };
  float* hW     = (float*)carve((size_t)N * HDIM * sizeof(float));
  float* agg    = (float*)carve((size_t)N * HDIM * sizeof(float));
  float* dinv   = (float*)carve((size_t)N * sizeof(float));
  float* nrm    = (float*)carve((size_t)E * sizeof(float));
  float* pooled = (float*)carve(64 * sizeof(float));

  int nB    = (N + 255) / 256;
  int eB    = (E + 255) / 256;
  int gemmB = (N + 63) / 64;  // 4 waves/block, 16 rows/wave
  int aggInitB = (int)(((long)N * 16 + 255) / 256);
  int aggEdgeB = (int)(((long)E * 16 + 255) / 256);
  int finB     = (int)(((long)N * HDIM + 255) / 256);

  // normalization precompute
  deg_init_kernel<<<nB, 256, 0, stream>>>(dinv, N);
  deg_acc_kernel <<<eB, 256, 0, stream>>>(row, edge_attr, dinv, E);
  dinv_kernel    <<<nB, 256, 0, stream>>>(dinv, N);
  norm_kernel    <<<eB, 256, 0, stream>>>(row, col, edge_attr, dinv, nrm, E);

  // layer 1: hW = x@W1 ; agg = selfloop + edge scatter ; relu(agg + b1)
  gemm_wmma_kernel<<<gemmB, 128, 0, stream>>>(x, W1, hW, N);
  agg_init_kernel <<<aggInitB, 256, 0, stream>>>(hW, dinv, agg, N);
  agg_edge_kernel <<<aggEdgeB, 256, 0, stream>>>(row, col, nrm, hW, agg, E);
  bias_relu_kernel<<<finB, 256, 0, stream>>>(agg, b1, N);

  // layer 2 (stream ordering makes buffer reuse safe: gemm consumes agg before init overwrites it)
  gemm_wmma_kernel<<<gemmB, 128, 0, stream>>>(agg, W2, hW, N);
  agg_init_kernel <<<aggInitB, 256, 0, stream>>>(hW, dinv, agg, N);
  agg_edge_kernel <<<aggEdgeB, 256, 0, stream>>>(row, col, nrm, hW, agg, E);
  bias_relu_kernel<<<finB, 256, 0, stream>>>(agg, b2, N);

  // mean pool + head
  pool_init_kernel<<<1, 64, 0, stream>>>(pooled);
  int rowsPerBlock = 4096;
  int poolB = (N + rowsPerBlock - 1) / rowsPerBlock;
  pool_kernel<<<poolB, 256, 0, stream>>>(agg, pooled, N, rowsPerBlock);
  head_kernel<<<1, 64, 0, stream>>>(pooled, h_other, Wc1, bc1, Wc2, bc2, (float*)d_out, N);
}